// FeaturePropagationLayer_43920335569147
// MI455X (gfx1250) — compile-verified
//
#include <hip/hip_runtime.h>

// ---------------- problem dims (fixed by the reference) ----------------
constexpr int B  = 8;
constexpr int N  = 8192;
constexpr int M  = 2048;
constexpr int C1 = 128;
constexpr int C2 = 256;
constexpr int CIN = C1 + C2;   // 384
constexpr int H0 = 256;
constexpr int H1 = 128;
constexpr int GROUPS = 32;
constexpr float GN_EPS = 1e-5f;
constexpr float NN_EPS = 1e-8f;

typedef __attribute__((ext_vector_type(2))) float v2f;
typedef __attribute__((ext_vector_type(8))) float v8f;

// =====================================================================
// Kernel 1: 3-NN search.  grid (N/256, B), block 256.
// xyz2 for the batch lives in LDS (2048*3*4 = 24 KB); reads broadcast.
// =====================================================================
__global__ void __launch_bounds__(256)
three_nn_kernel(const float* __restrict__ xyz1, const float* __restrict__ xyz2,
                int* __restrict__ idx3, float* __restrict__ w3) {
  __shared__ float sxyz[M * 3];
  const int b = blockIdx.y;
  const int n = blockIdx.x * 256 + threadIdx.x;
  const float* x2 = xyz2 + (size_t)b * M * 3;
  for (int i = threadIdx.x; i < M * 3; i += 256) sxyz[i] = x2[i];
  __syncthreads();

  const size_t pbase = ((size_t)b * N + n) * 3;
  const float px = xyz1[pbase + 0];
  const float py = xyz1[pbase + 1];
  const float pz = xyz1[pbase + 2];

  float d0 = 3.0e38f, d1 = 3.0e38f, d2 = 3.0e38f;
  int   i0 = 0,       i1 = 0,       i2 = 0;
  for (int m = 0; m < M; ++m) {
    const float dx = px - sxyz[m * 3 + 0];
    const float dy = py - sxyz[m * 3 + 1];
    const float dz = pz - sxyz[m * 3 + 2];
    const float d = dx * dx + dy * dy + dz * dz;
    if (d < d0)      { d2 = d1; i2 = i1; d1 = d0; i1 = i0; d0 = d; i0 = m; }
    else if (d < d1) { d2 = d1; i2 = i1; d1 = d;  i1 = m; }
    else if (d < d2) { d2 = d;  i2 = m; }
  }
  const float v0 = 1.0f / (d0 + NN_EPS);
  const float v1 = 1.0f / (d1 + NN_EPS);
  const float v2 = 1.0f / (d2 + NN_EPS);
  const float rs = 1.0f / (v0 + v1 + v2);
  idx3[pbase + 0] = i0;  w3[pbase + 0] = v0 * rs;
  idx3[pbase + 1] = i1;  w3[pbase + 1] = v1 * rs;
  idx3[pbase + 2] = i2;  w3[pbase + 2] = v2 * rs;
}

// =====================================================================
// Kernel 2: fused gather/interp/concat + GEMM1 (384 -> 256), WMMA f32.
// grid (N/32, B), block 256 = 8 waves.
// Block tile: 256 out-ch x 32 points; wave tile: 64 out-ch x 16 points
// (4 accumulators, 4 WMMAs per 5 ds_load_b64).
// K-pairs {k,k+1} are interleaved per column so every WMMA operand is a
// single aligned ds_load_b64 (no VGPR re-packing movs).
// =====================================================================
constexpr int KC1     = 12;          // K-chunk of w0 staged per iteration
constexpr int BPITCH1 = 66;          // floats per K-pair row of featT (32 pts * 2 + 2)
constexpr int APITCH1 = 546;         // floats per K-pair row of w0t   (256 o * 2 + 34)

__global__ void __launch_bounds__(256)
gemm1_fused_kernel(const float* __restrict__ points1, const float* __restrict__ points2,
                   const int* __restrict__ idx3, const float* __restrict__ w3,
                   const float* __restrict__ w0, float* __restrict__ h0) {
  __shared__ __align__(16) float ldsB[(CIN / 2) * BPITCH1];       // 50688 B
  __shared__ __align__(16) float ldsA[(KC1 / 2) * APITCH1];       // 13104 B
  __shared__ int   sIdx[32 * 3];
  __shared__ float sW[32 * 3];

  const int b   = blockIdx.y;
  const int n0  = blockIdx.x * 32;
  const int tid = threadIdx.x;

  if (tid < 96) {
    const int j = tid / 3, t = tid % 3;
    const size_t base = ((size_t)b * N + n0 + j) * 3 + t;
    sIdx[tid] = idx3[base];
    sW[tid]   = w3[base];
  }
  __syncthreads();

  // ---- build concat feature tile in LDS, pair-interleaved:
  //      featT pair p=c/2 row, element (j*2 + c&1) ----
  const float* p1 = points1 + ((size_t)b * N + n0) * C1;
  const float* p2 = points2 + (size_t)b * M * C2;
  for (int e = tid; e < 32 * CIN; e += 256) {
    const int j = e / CIN;
    const int c = e % CIN;                // consecutive threads -> consecutive c
    float v;
    if (c < C1) {
      v = p1[(size_t)j * C1 + c];
    } else {
      const int cc = c - C1;
      v = sW[j * 3 + 0] * p2[(size_t)sIdx[j * 3 + 0] * C2 + cc]
        + sW[j * 3 + 1] * p2[(size_t)sIdx[j * 3 + 1] * C2 + cc]
        + sW[j * 3 + 2] * p2[(size_t)sIdx[j * 3 + 2] * C2 + cc];
    }
    ldsB[(c >> 1) * BPITCH1 + j * 2 + (c & 1)] = v;
  }

  const int wave = tid >> 5;
  const int lane = tid & 31;
  const int hi   = lane >> 4;       // half-wave selects K rows {k,k+1} vs {k+2,k+3}
  const int mm   = lane & 15;
  const int ob   = (wave >> 1) * 64;   // 4 o-waves * 64 = 256 out-ch
  const int nb   = (wave & 1) * 16;    // 2 n-waves * 16 = 32 points

  v8f acc0 = {}, acc1 = {}, acc2 = {}, acc3 = {};

  for (int kc = 0; kc < CIN / KC1; ++kc) {
    __syncthreads();                  // feat tile ready / previous chunk consumed
    for (int e = tid; e < H0 * KC1; e += 256) {
      const int o = e / KC1;
      const int k = e % KC1;
      ldsA[(k >> 1) * APITCH1 + o * 2 + (k & 1)] = w0[(size_t)o * CIN + kc * KC1 + k];
      if (k == 0 && kc + 1 < CIN / KC1)
        __builtin_prefetch(&w0[(size_t)o * CIN + (kc + 1) * KC1], 0, 1);
    }
    __syncthreads();
#pragma unroll
    for (int kk = 0; kk < KC1; kk += 4) {
      const int bp = (kc * KC1 + kk) / 2 + hi;   // global K-pair row of featT
      const int ap = (kk >> 1) + hi;             // chunk-local K-pair row of w0t
      const v2f bv = *(const v2f*)(ldsB + bp * BPITCH1 + (nb + mm) * 2);
      const v2f a0 = *(const v2f*)(ldsA + ap * APITCH1 + (ob      + mm) * 2);
      const v2f a1 = *(const v2f*)(ldsA + ap * APITCH1 + (ob + 16 + mm) * 2);
      const v2f a2 = *(const v2f*)(ldsA + ap * APITCH1 + (ob + 32 + mm) * 2);
      const v2f a3 = *(const v2f*)(ldsA + ap * APITCH1 + (ob + 48 + mm) * 2);
      acc0 = __builtin_amdgcn_wmma_f32_16x16x4_f32(false, a0, false, bv, (short)0, acc0, false, false);
      acc1 = __builtin_amdgcn_wmma_f32_16x16x4_f32(false, a1, false, bv, (short)0, acc1, false, false);
      acc2 = __builtin_amdgcn_wmma_f32_16x16x4_f32(false, a2, false, bv, (short)0, acc2, false, false);
      acc3 = __builtin_amdgcn_wmma_f32_16x16x4_f32(false, a3, false, bv, (short)0, acc3, false, false);
    }
  }

  // D 16x16 f32: VGPR r holds M=r (lanes 0-15) and M=r+8 (lanes 16-31)
  float* out = h0 + (size_t)b * H0 * N;
  const int col = n0 + nb + mm;
#pragma unroll
  for (int r = 0; r < 8; ++r) {
    const int row = ob + r + 8 * hi;
    out[(size_t)(row)      * N + col] = acc0[r];
    out[(size_t)(row + 16) * N + col] = acc1[r];
    out[(size_t)(row + 32) * N + col] = acc2[r];
    out[(size_t)(row + 48) * N + col] = acc3[r];
  }
}

// =====================================================================
// Kernel 3/5: GroupNorm statistics. grid (32, B), block 256.
// =====================================================================
template <int CHAN, int CPG>
__global__ void __launch_bounds__(256)
gn_stats_kernel(const float* __restrict__ x, float* __restrict__ stats) {
  __shared__ float s1[256];
  __shared__ float s2[256];
  const int b = blockIdx.y, g = blockIdx.x;
  const float* xp = x + ((size_t)b * CHAN + (size_t)g * CPG) * N;
  float s = 0.0f, q = 0.0f;
  for (int i = threadIdx.x; i < CPG * N; i += 256) {
    const float v = xp[i];
    s += v;
    q += v * v;
  }
  s1[threadIdx.x] = s;
  s2[threadIdx.x] = q;
  __syncthreads();
  for (int st = 128; st > 0; st >>= 1) {
    if (threadIdx.x < st) {
      s1[threadIdx.x] += s1[threadIdx.x + st];
      s2[threadIdx.x] += s2[threadIdx.x + st];
    }
    __syncthreads();
  }
  if (threadIdx.x == 0) {
    const float cnt = (float)(CPG * N);
    const float mu  = s1[0] / cnt;
    const float var = s2[0] / cnt - mu * mu;   // population var == jnp.var
    stats[((size_t)b * GROUPS + g) * 2 + 0] = mu;
    stats[((size_t)b * GROUPS + g) * 2 + 1] = rsqrtf(var + GN_EPS);
  }
}

// =====================================================================
// Kernel 4: GN0 apply + ReLU, in place on h0 (B,256,N), float4 vectorized.
// =====================================================================
__global__ void __launch_bounds__(256)
gn0_apply_kernel(float* __restrict__ x, const float* __restrict__ stats,
                 const float* __restrict__ gamma, const float* __restrict__ beta) {
  const size_t idx = (size_t)blockIdx.x * 256 + threadIdx.x;   // float4 index
  const size_t total = (size_t)B * H0 * N / 4;
  if (idx >= total) return;
  const size_t e = idx * 4;
  const int c = (int)((e / N) % H0);
  const int b = (int)(e / ((size_t)N * H0));
  const int g = c >> 3;                                        // 8 ch / group
  const float mu = stats[((size_t)b * GROUPS + g) * 2 + 0];
  const float is = stats[((size_t)b * GROUPS + g) * 2 + 1];
  const float ga = gamma[c], be = beta[c];
  float4 v = ((const float4*)x)[idx];
  v.x = fmaxf((v.x - mu) * is * ga + be, 0.0f);
  v.y = fmaxf((v.y - mu) * is * ga + be, 0.0f);
  v.z = fmaxf((v.z - mu) * is * ga + be, 0.0f);
  v.w = fmaxf((v.w - mu) * is * ga + be, 0.0f);
  ((float4*)x)[idx] = v;
}

// =====================================================================
// Kernel 5b: GEMM2 (256 -> 128), WMMA f32. grid (N/32, B), block 256.
// Block tile: 128 out-ch x 32 points; wave tile: 32 out-ch x 16 points.
// Same pair-interleaved LDS layout as GEMM1.
// =====================================================================
constexpr int KC2     = 32;
constexpr int BPITCH2 = 66;          // 32 pts * 2 + 2
constexpr int APITCH2 = 290;         // 128 o * 2 + 34

__global__ void __launch_bounds__(256)
gemm2_kernel(const float* __restrict__ h0a, const float* __restrict__ w1,
             float* __restrict__ h1) {
  __shared__ __align__(16) float ldsB[(H0 / 2) * BPITCH2];   // 33792 B
  __shared__ __align__(16) float ldsA[(KC2 / 2) * APITCH2];  // 18560 B

  const int b   = blockIdx.y;
  const int n0  = blockIdx.x * 32;
  const int tid = threadIdx.x;

  const float* x = h0a + (size_t)b * H0 * N;
  for (int e = tid; e < H0 * 32; e += 256) {
    const int c = e >> 5;
    const int j = e & 31;
    ldsB[(c >> 1) * BPITCH2 + j * 2 + (c & 1)] = x[(size_t)c * N + n0 + j];
  }

  const int wave = tid >> 5;
  const int lane = tid & 31;
  const int hi   = lane >> 4;
  const int mm   = lane & 15;
  const int ob   = (wave >> 1) * 32;   // 4 o-waves * 32 = 128 out-ch
  const int nb   = (wave & 1) * 16;

  v8f acc0 = {}, acc1 = {};
  for (int kc = 0; kc < H0 / KC2; ++kc) {
    __syncthreads();
    for (int e = tid; e < H1 * KC2; e += 256) {
      const int o = e >> 5;
      const int k = e & 31;
      ldsA[(k >> 1) * APITCH2 + o * 2 + (k & 1)] = w1[(size_t)o * H0 + kc * KC2 + k];
      if (k == 0 && kc + 1 < H0 / KC2)
        __builtin_prefetch(&w1[(size_t)o * H0 + (kc + 1) * KC2], 0, 1);
    }
    __syncthreads();
#pragma unroll
    for (int kk = 0; kk < KC2; kk += 4) {
      const int bp = (kc * KC2 + kk) / 2 + hi;
      const int ap = (kk >> 1) + hi;
      const v2f bv = *(const v2f*)(ldsB + bp * BPITCH2 + (nb + mm) * 2);
      const v2f a0 = *(const v2f*)(ldsA + ap * APITCH2 + (ob      + mm) * 2);
      const v2f a1 = *(const v2f*)(ldsA + ap * APITCH2 + (ob + 16 + mm) * 2);
      acc0 = __builtin_amdgcn_wmma_f32_16x16x4_f32(false, a0, false, bv, (short)0, acc0, false, false);
      acc1 = __builtin_amdgcn_wmma_f32_16x16x4_f32(false, a1, false, bv, (short)0, acc1, false, false);
    }
  }

  float* out = h1 + (size_t)b * H1 * N;
  const int col = n0 + nb + mm;
#pragma unroll
  for (int r = 0; r < 8; ++r) {
    const int row = ob + r + 8 * hi;
    out[(size_t)(row)      * N + col] = acc0[r];
    out[(size_t)(row + 16) * N + col] = acc1[r];
  }
}

// =====================================================================
// Kernel 6: GN1 apply + ReLU + transpose (B,128,N) -> (B,N,128).
// grid (N/32, 128/32, B), block 256; 32x32 LDS tile.
// =====================================================================
__global__ void __launch_bounds__(256)
gn1_apply_transpose_kernel(const float* __restrict__ h1, const float* __restrict__ stats,
                           const float* __restrict__ gamma, const float* __restrict__ beta,
                           float* __restrict__ out) {
  __shared__ float tile[32][33];
  const int b  = blockIdx.z;
  const int c0 = blockIdx.y * 32;
  const int n0 = blockIdx.x * 32;
  const int tid = threadIdx.x;
  const float* x = h1 + (size_t)b * H1 * N;

#pragma unroll
  for (int i = 0; i < 4; ++i) {
    const int e  = tid + i * 256;
    const int ci = e >> 5;
    const int nj = e & 31;
    const int c  = c0 + ci;
    const int g  = c >> 2;                                   // 4 ch / group
    const float mu = stats[((size_t)b * GROUPS + g) * 2 + 0];
    const float is = stats[((size_t)b * GROUPS + g) * 2 + 1];
    float v = x[(size_t)c * N + n0 + nj];
    v = fmaxf((v - mu) * is * gamma[c] + beta[c], 0.0f);
    tile[ci][nj] = v;
  }
  __syncthreads();
#pragma unroll
  for (int i = 0; i < 4; ++i) {
    const int e  = tid + i * 256;
    const int nj = e >> 5;
    const int ci = e & 31;
    out[(((size_t)b * N) + n0 + nj) * H1 + c0 + ci] = tile[ci][nj];
  }
}

// =====================================================================
// Launcher
// =====================================================================
extern "C" void kernel_launch(void* const* d_in, const int* in_sizes, int n_in,
                              void* d_out, int out_size, void* d_ws, size_t ws_size,
                              hipStream_t stream) {
  (void)in_sizes; (void)n_in; (void)out_size; (void)ws_size;
  const float* xyz1    = (const float*)d_in[0];
  const float* xyz2    = (const float*)d_in[1];
  const float* points1 = (const float*)d_in[2];
  const float* points2 = (const float*)d_in[3];
  const float* w0      = (const float*)d_in[4];
  const float* w1      = (const float*)d_in[5];
  const float* gn0_w   = (const float*)d_in[6];
  const float* gn0_b   = (const float*)d_in[7];
  const float* gn1_w   = (const float*)d_in[8];
  const float* gn1_b   = (const float*)d_in[9];
  float* out = (float*)d_out;

  char* ws = (char*)d_ws;
  size_t off = 0;
  auto wsAlloc = [&](size_t bytes) -> void* {
    void* p = ws + off;
    off = (off + bytes + 255) & ~(size_t)255;
    return p;
  };
  int*   idx3 = (int*)  wsAlloc(sizeof(int)   * (size_t)B * N * 3);
  float* w3   = (float*)wsAlloc(sizeof(float) * (size_t)B * N * 3);
  float* h0   = (float*)wsAlloc(sizeof(float) * (size_t)B * H0 * N);
  float* h1   = (float*)wsAlloc(sizeof(float) * (size_t)B * H1 * N);
  float* st0  = (float*)wsAlloc(sizeof(float) * (size_t)B * GROUPS * 2);
  float* st1  = (float*)wsAlloc(sizeof(float) * (size_t)B * GROUPS * 2);

  three_nn_kernel<<<dim3(N / 256, B), 256, 0, stream>>>(xyz1, xyz2, idx3, w3);
  gemm1_fused_kernel<<<dim3(N / 32, B), 256, 0, stream>>>(points1, points2, idx3, w3, w0, h0);
  gn_stats_kernel<H0, H0 / GROUPS><<<dim3(GROUPS, B), 256, 0, stream>>>(h0, st0);
  {
    const size_t tot4 = (size_t)B * H0 * N / 4;
    gn0_apply_kernel<<<(unsigned)((tot4 + 255) / 256), 256, 0, stream>>>(h0, st0, gn0_w, gn0_b);
  }
  gemm2_kernel<<<dim3(N / 32, B), 256, 0, stream>>>(h0, w1, h1);
  gn_stats_kernel<H1, H1 / GROUPS><<<dim3(GROUPS, B), 256, 0, stream>>>(h1, st1);
  gn1_apply_transpose_kernel<<<dim3(N / 32, H1 / 32, B), 256, 0, stream>>>(h1, st1, gn1_w, gn1_b, out);
}